// Conv1D_local_39058432590491
// MI455X (gfx1250) — compile-verified
//
#include <hip/hip_runtime.h>
#include <math.h>

// Problem constants (match reference)
#define B_  32
#define L_  32768
#define C_  128
#define K_  64
#define PAD_LO 31          // XLA SAME, stride 1, K even: low = (K-1)/2 = 31

// Tiling
#define CH_TILE   32       // channels per block (one lane per channel within a wave)
#define POS_TILE  256      // output positions per block
#define ROWS_LOAD 320      // rows staged: POS_TILE + K - 1 = 319, padded to 320 (5 rounds)
#define ROWS_ALLOC 324     // + slack rows for unconditional window refills (max row 320)
#define NTHREADS  512      // (32, 16)

typedef float v2f __attribute__((ext_vector_type(2)));

// ---- CDNA5 async global->LDS path (probe-confirmed signature: int4 AS1* / AS3*) ----
typedef int v4i_cdna5 __attribute__((ext_vector_type(4)));
#define TO_LDS(p) ((__attribute__((address_space(3))) v4i_cdna5*)(unsigned int)(unsigned long long)(p))
#define TO_GLB(p) ((__attribute__((address_space(1))) v4i_cdna5*)(unsigned long long)(p))

#if defined(__has_builtin)
#  if __has_builtin(__builtin_amdgcn_global_load_async_to_lds_b128)
#    define ASYNC_COPY16(gp, lp) \
        __builtin_amdgcn_global_load_async_to_lds_b128(TO_GLB(gp), TO_LDS(lp), 0, 0)
#    if __has_builtin(__builtin_amdgcn_s_wait_asynccnt)
#      define ASYNC_WAIT() __builtin_amdgcn_s_wait_asynccnt(0)
#    else
#      define ASYNC_WAIT() asm volatile("s_wait_asynccnt 0" ::: "memory")
#    endif
#  endif
#  if __has_builtin(__builtin_amdgcn_tanhf)
#    define TANH_F32(v) __builtin_amdgcn_tanhf(v)
#  endif
#endif

#ifndef ASYNC_COPY16
#  define ASYNC_COPY16(gp, lp) (*(float4*)(lp) = *(const float4*)(gp))
#  define ASYNC_WAIT() ((void)0)
#endif
#ifndef TANH_F32
#  define TANH_F32(v) tanhf(v)
#endif

// ---- Forced V_PK_FMA_F32 (VOP3P op 31): 2 f32 FMAs/instr, full-rate on CDNA ----
// src1 word-broadcast via op_sel/op_sel_hi: one {w_k, w_k+1} pair feeds both taps.
#if defined(__gfx1250__)
__device__ __forceinline__ void pk_fma_wlo(v2f& acc, v2f xv, v2f wp) {
    // both halves of src1 take word 0 (w_k)
    asm("v_pk_fma_f32 %0, %1, %2, %0 op_sel:[0,0,0] op_sel_hi:[1,0,1]"
        : "+v"(acc) : "v"(xv), "v"(wp));
}
__device__ __forceinline__ void pk_fma_whi(v2f& acc, v2f xv, v2f wp) {
    // both halves of src1 take word 1 (w_k+1)
    asm("v_pk_fma_f32 %0, %1, %2, %0 op_sel:[0,1,0] op_sel_hi:[1,1,1]"
        : "+v"(acc) : "v"(xv), "v"(wp));
}
#else
__device__ __forceinline__ void pk_fma_wlo(v2f& acc, v2f xv, v2f wp) {
    acc = __builtin_elementwise_fma(xv, (v2f){wp.x, wp.x}, acc);
}
__device__ __forceinline__ void pk_fma_whi(v2f& acc, v2f xv, v2f wp) {
    acc = __builtin_elementwise_fma(xv, (v2f){wp.y, wp.y}, acc);
}
#endif

__global__ __launch_bounds__(NTHREADS)
void dwconv1d_tanh_kernel(const float* __restrict__ x,     // (B, L, C)
                          const float* __restrict__ w,     // (K, 1, C) -> k*C + c
                          const float* __restrict__ bias,  // (C,)
                          float* __restrict__ out)         // (B, L, C)
{
    __shared__ __align__(16) float xs[ROWS_ALLOC * CH_TILE]; // [row][ch], ~40.5 KB
    __shared__ __align__(16) float ws[K_ * CH_TILE];         // [tap][ch],  8 KB

    const int tx  = threadIdx.x;                 // channel lane: 0..31
    const int ty  = threadIdx.y;                 // position group: 0..15 (one wave each)
    const int tid = ty * CH_TILE + tx;

    const int tile = blockIdx.x;                 // 0..127  (L / POS_TILE)
    const int cg   = blockIdx.y;                 // 0..3    (C / CH_TILE)
    const int b    = blockIdx.z;                 // 0..31

    const int l0 = tile * POS_TILE;
    const int c0 = cg * CH_TILE;

    // ------------- stage x halo tile + weights into LDS (async DMA) -------------
    {
        const int q  = (tid & 7) * 4;            // float offset of this 16B chunk in a row
        const int r0 = tid >> 3;                 // starting row (0..63); 512 thr cover 64 rows

        #pragma unroll
        for (int rr = 0; rr < ROWS_LOAD / 64; ++rr) { // 5 rounds -> rows 0..319
            const int r = r0 + rr * 64;
            const int l = l0 - PAD_LO + r;       // global position of this row
            float* lp = &xs[r * CH_TILE + q];
            if ((unsigned)l < (unsigned)L_) {
                const float* gp = x + ((size_t)b * L_ + l) * C_ + c0 + q;
                ASYNC_COPY16(gp, lp);
            } else {
                *(float4*)lp = make_float4(0.f, 0.f, 0.f, 0.f);  // zero padding
            }
        }
        {   // weights: 64 rows x 128B = one round for 512 threads
            const int k = r0;
            const float* gp = w + (size_t)k * C_ + c0 + q;
            ASYNC_COPY16(gp, &ws[k * CH_TILE + q]);
        }
        ASYNC_WAIT();
    }
    __syncthreads();

    // ------------- compute: 16 consecutive outputs per thread, one channel -------------
    // out[l0+base+t] = sum_k x[l0+base+t+k-31] * w[k]  ; LDS row index = base + t + k
    // Two shifted float2 sliding windows so every tap is an aligned packed FMA:
    //   phase s = k/2 (k even):
    //   e[(s+j)&7] = rows (base+k+2j, base+k+2j+1)   -> pairs with tap k
    //   o[(s+j)&7] = rows (base+k+2j+1, base+k+2j+2) -> pairs with tap k+1
    const int base = ty * 16;                    // 0,16,...,240

    v2f acc2[8];
    v2f e[8], o[8];
    #pragma unroll
    for (int j = 0; j < 8; ++j) acc2[j] = (v2f){0.f, 0.f};
    #pragma unroll
    for (int j = 0; j < 8; ++j) {
        const float r0v = xs[(base + 2*j    ) * CH_TILE + tx];
        const float r1v = xs[(base + 2*j + 1) * CH_TILE + tx];
        const float r2v = xs[(base + 2*j + 2) * CH_TILE + tx];
        e[j] = (v2f){r0v, r1v};
        o[j] = (v2f){r1v, r2v};
    }

    for (int kk = 0; kk < K_; kk += 16) {        // 4 iterations; slot phase repeats per 16 taps
        #pragma unroll
        for (int s2 = 0; s2 < 8; ++s2) {         // one tap pair (k, k+1) per step
            const int k = kk + 2 * s2;
            const v2f wp = (v2f){ ws[(k    ) * CH_TILE + tx],
                                  ws[(k + 1) * CH_TILE + tx] };   // ds_load_2addr pair
            #pragma unroll
            for (int j = 0; j < 8; ++j) {
                const int s = (s2 + j) & 7;
                pk_fma_wlo(acc2[j], e[s], wp);   // tap k   : v_pk_fma_f32 (w lo bcast)
                pk_fma_whi(acc2[j], o[s], wp);   // tap k+1 : v_pk_fma_f32 (w hi bcast)
            }
            // advance both windows: refill slot s2 with the next row pair
            const int rtop = base + k + 16;      // <= 318 real; +2 slack -> 320 max (in-bounds)
            const float a0 = xs[(rtop    ) * CH_TILE + tx];
            const float a1 = xs[(rtop + 1) * CH_TILE + tx];
            const float a2 = xs[(rtop + 2) * CH_TILE + tx];
            e[s2] = (v2f){a0, a1};
            o[s2] = (v2f){a1, a2};
        }
    }

    // ------------- bias + tanh + NT store -------------
    const float bv = bias[c0 + tx];
    float* op = out + ((size_t)b * L_ + l0 + base) * C_ + c0 + tx;
    #pragma unroll
    for (int j = 0; j < 8; ++j) {
        float v0 = TANH_F32(acc2[j].x + bv);
        float v1 = TANH_F32(acc2[j].y + bv);
        __builtin_nontemporal_store(v0, op + (size_t)(2*j    ) * C_);
        __builtin_nontemporal_store(v1, op + (size_t)(2*j + 1) * C_);
    }
}

extern "C" void kernel_launch(void* const* d_in, const int* in_sizes, int n_in,
                              void* d_out, int out_size, void* d_ws, size_t ws_size,
                              hipStream_t stream) {
    (void)in_sizes; (void)n_in; (void)d_ws; (void)ws_size; (void)out_size;
    const float* x    = (const float*)d_in[0];
    const float* w    = (const float*)d_in[1];
    const float* bias = (const float*)d_in[2];
    float* out        = (float*)d_out;

    dim3 grid(L_ / POS_TILE, C_ / CH_TILE, B_);   // (128, 4, 32)
    dim3 block(CH_TILE, NTHREADS / CH_TILE);      // (32, 16)
    hipLaunchKernelGGL(dwconv1d_tanh_kernel, grid, block, 0, stream, x, w, bias, out);
}